// MultiSparkMemoryNet_19997367730507
// MI455X (gfx1250) — compile-verified
//
#include <hip/hip_runtime.h>
#include <math.h>

#define NN 8192
#define SS 256

#define BRANCH_PROB 0.3f
#define BRANCH_THRESHOLD 0.6f
#define LR_PATH 0.02f
#define NOISE_STD 0.05f
#define SPARK_FORCE_STEPS 5
#define DECAY_VIS (1.0f - 0.001f * 0.02f)   /* 0.99998 */
#define DECAY_NOVIS (1.0f - 0.001f)         /* 0.999   */

typedef __attribute__((ext_vector_type(2))) float v2f;
typedef __attribute__((ext_vector_type(8))) float v8f;

// ---------------------------------------------------------------------------
// K1: zero workspace (row_flag[NN] + main_next[SS])
// ---------------------------------------------------------------------------
__global__ void k_zero_ws(unsigned int* ws, int n) {
    int i = blockIdx.x * blockDim.x + threadIdx.x;
    if (i < n) ws[i] = 0u;
}

// ---------------------------------------------------------------------------
// K2: s_out = sigmoid(W @ (0.95*s) + 0.05*noise) via V_WMMA_F32_16X16X4_F32.
// One block (8 waves) per 16-row tile; wave w covers K in [w*1024, w*1024+1024).
// A (16x4 f32): lanes 0-15 = M, VGPR0 holds K=koff, VGPR1 K=koff+1,
// koff = (lane>>4)*2.  B (4x16): 0.95*s[k] replicated across all N columns, so
// every column of C holds the dot product.  C/D: lane0 -> M=r, lane16 -> M=8+r.
// ---------------------------------------------------------------------------
__global__ void k_gemv_wmma(const float* __restrict__ W,
                            const float* __restrict__ s,
                            const float* __restrict__ noise,
                            float* __restrict__ s_out) {
    const int m0   = blockIdx.x * 16;
    const int tid  = threadIdx.x;
    const int wave = tid >> 5;
    const int lane = tid & 31;
    const int rowl = lane & 15;
    const int koff = (lane >> 4) * 2;

    const float* __restrict__ wrow = W + (size_t)(m0 + rowl) * NN;
    const int kbeg = wave * 1024;
    const int kend = kbeg + 1024;

    v8f acc = {};
    for (int kc = kbeg; kc < kend; kc += 4) {
        v2f a, b;
        a.x = wrow[kc + koff];
        a.y = wrow[kc + koff + 1];
        b.x = 0.95f * s[kc + koff];
        b.y = 0.95f * s[kc + koff + 1];
        acc = __builtin_amdgcn_wmma_f32_16x16x4_f32(
                  /*neg_a=*/false, a, /*neg_b=*/false, b,
                  /*c_mod=*/(short)0, acc, /*reuse_a=*/false, /*reuse_b=*/false);
    }

    __shared__ float part[8 * 16];
    if (lane == 0) {
#pragma unroll
        for (int r = 0; r < 8; ++r) part[wave * 16 + r] = acc[r];
    }
    if (lane == 16) {
#pragma unroll
        for (int r = 0; r < 8; ++r) part[wave * 16 + 8 + r] = acc[r];
    }
    __syncthreads();

    if (tid < 16) {
        float v = 0.0f;
#pragma unroll
        for (int w = 0; w < 8; ++w) v += part[w * 16 + tid];
        v += NOISE_STD * noise[m0 + tid];
        s_out[m0 + tid] = 1.0f / (1.0f + __expf(-v));
    }
}

// ---------------------------------------------------------------------------
// K3: force young active sparks' positions to 1.0 (all racers write 1.0)
// ---------------------------------------------------------------------------
__global__ void k_force(const int* __restrict__ pos,
                        const int* __restrict__ age,
                        const unsigned char* __restrict__ active,
                        float* __restrict__ s_out) {
    int t = blockIdx.x * blockDim.x + threadIdx.x;
    if (t < SS && active[t] && age[t] < SPARK_FORCE_STEPS)
        s_out[pos[t]] = 1.0f;
}

// ---------------------------------------------------------------------------
// K4: per-spark (1 block/spark): zero delta & hit rows in d_out, compute
// main_next by inverse-CDF count over positive = relu(W[pos,:]) + 1e-6,
// mark row_flag.  count = #{k : inclusive_prefix(positive)[k] < u*total}.
// ---------------------------------------------------------------------------
__global__ void k_spark_prep(const float* __restrict__ W,
                             const int* __restrict__ pos,
                             const float* __restrict__ u_main,
                             float* __restrict__ outW,
                             float* __restrict__ outV,
                             int* __restrict__ main_next,
                             unsigned int* __restrict__ row_flag) {
    const int sidx = blockIdx.x;
    const int t = threadIdx.x;                    // 256 threads
    const int p = pos[sidx];
    const float* __restrict__ row = W + (size_t)p * NN;
    float4* drow = (float4*)(outW + (size_t)p * NN);
    float4* vrow = (float4*)(outV + (size_t)p * NN);

    const float4 z4 = make_float4(0.f, 0.f, 0.f, 0.f);
    for (int j = t; j < NN / 4; j += 256) { drow[j] = z4; vrow[j] = z4; }

    const int base = t * 32;                      // 8192/256 contiguous elems
    float lsum = 0.0f;
#pragma unroll 8
    for (int j = 0; j < 32; ++j)
        lsum += fmaxf(row[base + j], 0.0f) + 1e-6f;

    __shared__ float ssum[256];
    __shared__ float sexcl[256];
    __shared__ float stotal;
    ssum[t] = lsum;
    __syncthreads();
    if (t == 0) {
        float run = 0.0f;
        for (int k = 0; k < 256; ++k) { sexcl[k] = run; run += ssum[k]; }
        stotal = run;
    }
    __syncthreads();

    const float target = u_main[sidx] * stotal;
    float running = sexcl[t];
    int cnt = 0;
#pragma unroll 8
    for (int j = 0; j < 32; ++j) {
        running += fmaxf(row[base + j], 0.0f) + 1e-6f;
        cnt += (running < target) ? 1 : 0;
    }

    __shared__ int scnt[256];
    scnt[t] = cnt;
    __syncthreads();
    for (int off = 128; off > 0; off >>= 1) {
        if (t < off) scnt[t] += scnt[t + off];
        __syncthreads();
    }
    if (t == 0) {
        int mn = scnt[0];
        if (mn > NN - 1) mn = NN - 1;
        main_next[sidx] = mn;
        row_flag[p] = 1u;
    }
}

// ---------------------------------------------------------------------------
// K5: scatter path reinforcement.  counts = onehot(main_next) + branch;
// atomicAdd deltas into the (zeroed) spark rows of d_out W; mark hits (1.0f).
// ---------------------------------------------------------------------------
__global__ void k_spark_scatter(const float* __restrict__ W,
                                const int* __restrict__ pos,
                                const float* __restrict__ energy,
                                const unsigned char* __restrict__ active,
                                const float* __restrict__ u_branch,
                                const int* __restrict__ main_next,
                                float* __restrict__ outW,
                                float* __restrict__ outV) {
    const int sidx = blockIdx.x;
    if (!active[sidx]) return;
    const int t = threadIdx.x;
    const int p = pos[sidx];
    const int mn = main_next[sidx];
    const float scale = LR_PATH * energy[sidx];
    const float* __restrict__ row = W + (size_t)p * NN;
    const float* __restrict__ ub  = u_branch + (size_t)sidx * NN;
    float* drow = outW + (size_t)p * NN;
    float* vrow = outV + (size_t)p * NN;

    for (int n = t; n < NN; n += 256) {
        float posv = fmaxf(row[n], 0.0f) + 1e-6f;
        int add = ((posv > BRANCH_THRESHOLD) && (ub[n] < BRANCH_PROB)) ? 1 : 0;
        add += (n == mn) ? 1 : 0;
        if (add) {
            atomicAdd(&drow[n], scale * (float)add);
            vrow[n] = 1.0f;
        }
    }
}

// ---------------------------------------------------------------------------
// K6: fused decay + clip + visited output, float4-vectorized grid-stride.
// Delta/hit regions are only read for flagged rows (zeroed fresh each call).
// ---------------------------------------------------------------------------
__global__ void k_decay(const float* __restrict__ W_in,
                        const unsigned char* __restrict__ vis_in,
                        const unsigned int* __restrict__ row_flag,
                        float* __restrict__ outW,
                        float* __restrict__ outV) {
    const size_t total4 = (size_t)NN * NN / 4;
    const size_t stride = (size_t)gridDim.x * blockDim.x;
    for (size_t i4 = (size_t)blockIdx.x * blockDim.x + threadIdx.x;
         i4 < total4; i4 += stride) {
        const size_t i = i4 * 4;
        const int r = (int)(i / NN);              // NN%4==0: one row per vec4
        const bool f = row_flag[r] != 0u;

        float4 w = ((const float4*)W_in)[i4];
        uchar4 v = ((const uchar4*)vis_in)[i4];
        float4 d = make_float4(0.f, 0.f, 0.f, 0.f);
        float4 h = make_float4(0.f, 0.f, 0.f, 0.f);
        if (f) { d = ((float4*)outW)[i4]; h = ((float4*)outV)[i4]; }

        float wv[4] = {w.x + d.x, w.y + d.y, w.z + d.z, w.w + d.w};
        unsigned char vb[4] = {v.x, v.y, v.z, v.w};
        float hb[4] = {h.x, h.y, h.z, h.w};
        float ow[4], ov[4];
#pragma unroll
        for (int c = 0; c < 4; ++c) {
            bool vis = (vb[c] != 0) || (hb[c] != 0.0f);
            float dec = vis ? DECAY_VIS : DECAY_NOVIS;
            float x = wv[c] * dec;
            x = fminf(fmaxf(x, -2.0f), 2.0f);
            ow[c] = x;
            ov[c] = vis ? 1.0f : 0.0f;
        }
        ((float4*)outW)[i4] = make_float4(ow[0], ow[1], ow[2], ow[3]);
        ((float4*)outV)[i4] = make_float4(ov[0], ov[1], ov[2], ov[3]);
    }
}

// ---------------------------------------------------------------------------
extern "C" void kernel_launch(void* const* d_in, const int* in_sizes, int n_in,
                              void* d_out, int out_size, void* d_ws, size_t ws_size,
                              hipStream_t stream) {
    (void)in_sizes; (void)n_in; (void)out_size; (void)ws_size;

    const float*         W        = (const float*)d_in[0];
    const float*         s        = (const float*)d_in[1];
    const unsigned char* visited  = (const unsigned char*)d_in[2];
    const float*         noise    = (const float*)d_in[3];
    const int*           pos      = (const int*)d_in[4];
    const float*         energy   = (const float*)d_in[5];
    const int*           age      = (const int*)d_in[6];
    const unsigned char* active   = (const unsigned char*)d_in[7];
    const float*         u_main   = (const float*)d_in[8];
    const float*         u_branch = (const float*)d_in[9];

    float* outW = (float*)d_out;                       // [NN*NN]
    float* outS = outW + (size_t)NN * NN;              // [NN]
    float* outV = outS + NN;                           // [NN*NN] as 0/1 f32

    unsigned int* row_flag  = (unsigned int*)d_ws;     // [NN]
    int*          main_next = (int*)d_ws + NN;         // [SS]

    // K1: clear flags/scratch
    k_zero_ws<<<(NN + SS + 255) / 256, 256, 0, stream>>>((unsigned int*)d_ws, NN + SS);

    // K2: WMMA f32 GEMV + sigmoid
    k_gemv_wmma<<<NN / 16, 256, 0, stream>>>(W, s, noise, outS);

    // K3: force young active sparks
    k_force<<<1, SS, 0, stream>>>(pos, age, active, outS);

    // K4: per-spark sampling + row zero + flags
    k_spark_prep<<<SS, 256, 0, stream>>>(W, pos, u_main, outW, outV, main_next, row_flag);

    // K5: scatter reinforcement deltas + visited hits
    k_spark_scatter<<<SS, 256, 0, stream>>>(W, pos, energy, active, u_branch,
                                            main_next, outW, outV);

    // K6: fused decay + clip + visited materialization
    k_decay<<<4096, 256, 0, stream>>>(W, visited, row_flag, outW, outV);
}